// recurrent_A3TGCN_54202487275564
// MI455X (gfx1250) — compile-verified
//
#include <hip/hip_runtime.h>
#include <math.h>

// ---------------------------------------------------------------------------
// A3TGCN forward (PERIODS=1, H0=0) on MI455X / gfx1250, wave32 + WMMA f32.
// Fused pipeline:
//   1) Xt = relu(x @ fc0_w + b)                      (WMMA GEMM)
//   2) dinv = rsqrt(1 + scatter(ew))                 (atomics)
//   3) {hz,hh} = Xt @ {Wz,Wh}; agg = b + h*dinv^2    (dual-WMMA GEMM, fused epilogue)
//   4) agg[dst] += h[src]*norm over edges            (atomics)
//   5) Z=sigmoid(aggZ@Lz+b); T=tanh(aggH@Lh+b);
//      h=(1-Z)*T -> out_h; out += relu(h)@fc_w       (dual-WMMA GEMM, fused head)
// ---------------------------------------------------------------------------

typedef __attribute__((ext_vector_type(2))) float v2f;
typedef __attribute__((ext_vector_type(8))) float v8f;

#define ACT_NONE    0
#define ACT_RELU    1

static __device__ __forceinline__ v8f wmma4(v2f a, v2f b, v8f c)
{
    return __builtin_amdgcn_wmma_f32_16x16x4_f32(
        /*neg_a=*/false, a, /*neg_b=*/false, b,
        /*c_mod=*/(short)0, c, /*reuse_a=*/false, /*reuse_b=*/false);
}

// ---------------------------------------------------------------------------
// Generic fp32 GEMM:  C[N x M] = act(A[N x K] @ W[K x M] + bias)
// block = 128 threads = 4 waves; each wave: one 16x16 tile of C.
// gridDim.x = ceil(N/16), gridDim.y = M/64.
// ---------------------------------------------------------------------------
__global__ __launch_bounds__(128) void wmma_gemm_f32(
    const float* __restrict__ A, const float* __restrict__ W,
    const float* __restrict__ bias, float* __restrict__ C,
    int N, int K, int M, int act)
{
    const int wave    = threadIdx.x >> 5;
    const int lane    = threadIdx.x & 31;
    const int rowTile = blockIdx.x;
    const int colTile = blockIdx.y * 4 + wave;

    const int m    = lane & 15;
    const int half = lane >> 4;

    int row = rowTile * 16 + m;
    if (row >= N) row = N - 1;            // clamp: EXEC stays all-ones for WMMA
    const int col = colTile * 16 + m;

    const float* aPtr = A + (size_t)row * K + (half << 1);
    const float* bPtr = W + (size_t)(half << 1) * M + col;

    v8f acc = {};
    for (int k = 0; k < K; k += 4) {
        v2f a, b;
        a.x = aPtr[0];  a.y = aPtr[1];
        b.x = bPtr[0];  b.y = bPtr[M];
        acc = wmma4(a, b, acc);
        aPtr += 4;
        bPtr += (size_t)4 * M;
    }

    const float bv = bias ? bias[col] : 0.0f;
#pragma unroll
    for (int i = 0; i < 8; ++i) {
        const int r = rowTile * 16 + i + half * 8;
        if (r < N) {
            float v = acc[i] + bv;
            if (act == ACT_RELU) v = fmaxf(v, 0.0f);
            C[(size_t)r * M + col] = v;
        }
    }
}

// ---------------------------------------------------------------------------
// Fused conv GEMMs: hz = Xt@Wz, hh = Xt@Wh (shared A fragment), plus the
// GCN self-loop/bias epilogue: agg* = b* + h*[r]*dinv[r]^2.
// K = 256, M = 128.  gridDim = (ceil(N/16), 2), block = 128.
// ---------------------------------------------------------------------------
__global__ __launch_bounds__(128) void conv_dual_gemm(
    const float* __restrict__ A,
    const float* __restrict__ Wz, const float* __restrict__ Wh,
    const float* __restrict__ bz, const float* __restrict__ bh,
    const float* __restrict__ dinv,
    float* __restrict__ hz, float* __restrict__ hh,
    float* __restrict__ aggZ, float* __restrict__ aggH, int N)
{
    const int K = 256, M = 128;
    const int wave    = threadIdx.x >> 5;
    const int lane    = threadIdx.x & 31;
    const int rowTile = blockIdx.x;
    const int colTile = blockIdx.y * 4 + wave;

    const int m    = lane & 15;
    const int half = lane >> 4;

    int row = rowTile * 16 + m;
    if (row >= N) row = N - 1;
    const int col = colTile * 16 + m;

    const float* aPtr  = A  + (size_t)row * K + (half << 1);
    const float* bzPtr = Wz + (size_t)(half << 1) * M + col;
    const float* bhPtr = Wh + (size_t)(half << 1) * M + col;

    v8f accZ = {};
    v8f accH = {};
    for (int k = 0; k < K; k += 4) {
        v2f a, fz, fh;
        a.x  = aPtr[0];   a.y  = aPtr[1];
        fz.x = bzPtr[0];  fz.y = bzPtr[M];
        fh.x = bhPtr[0];  fh.y = bhPtr[M];
        accZ = wmma4(a, fz, accZ);
        accH = wmma4(a, fh, accH);
        aPtr  += 4;
        bzPtr += (size_t)4 * M;
        bhPtr += (size_t)4 * M;
    }

    const float bzv = bz[col];
    const float bhv = bh[col];
#pragma unroll
    for (int i = 0; i < 8; ++i) {
        const int r = rowTile * 16 + i + half * 8;
        if (r < N) {
            float w = dinv[r];
            w = w * w;
            const float vz = accZ[i];
            const float vh = accH[i];
            const size_t o = (size_t)r * M + col;
            hz[o]   = vz;
            hh[o]   = vh;
            aggZ[o] = bzv + vz * w;   // bias + self-loop contribution
            aggH[o] = bhv + vh * w;
        }
    }
}

// ---------------------------------------------------------------------------
// Fused lin GEMMs + GRU gate + output head.
// Z = sigmoid(aggZ @ Lz + bz), T = tanh(aggH @ Lh + bh), h = (1-Z)*T.
// h -> outh; partial dot relu(h).fc_w reduced across each half-wave (one row
// per 16 lanes) and atomically accumulated into outv (pre-init to fc_b).
// K = M = 128.  gridDim = (ceil(N/16), 2), block = 128.
// ---------------------------------------------------------------------------
__global__ __launch_bounds__(128) void lin_gate_head(
    const float* __restrict__ aggZ, const float* __restrict__ aggH,
    const float* __restrict__ Lz, const float* __restrict__ Lh,
    const float* __restrict__ bz, const float* __restrict__ bh,
    const float* __restrict__ fc_w,
    float* __restrict__ outv, float* __restrict__ outh, int N)
{
    const int K = 128, M = 128;
    const int wave    = threadIdx.x >> 5;
    const int lane    = threadIdx.x & 31;
    const int rowTile = blockIdx.x;
    const int colTile = blockIdx.y * 4 + wave;

    const int m    = lane & 15;
    const int half = lane >> 4;

    int row = rowTile * 16 + m;
    if (row >= N) row = N - 1;
    const int col = colTile * 16 + m;

    const float* azPtr = aggZ + (size_t)row * K + (half << 1);
    const float* ahPtr = aggH + (size_t)row * K + (half << 1);
    const float* bzPtr = Lz + (size_t)(half << 1) * M + col;
    const float* bhPtr = Lh + (size_t)(half << 1) * M + col;

    v8f accZ = {};
    v8f accH = {};
    for (int k = 0; k < K; k += 4) {
        v2f az, ah, fz, fh;
        az.x = azPtr[0];  az.y = azPtr[1];
        ah.x = ahPtr[0];  ah.y = ahPtr[1];
        fz.x = bzPtr[0];  fz.y = bzPtr[M];
        fh.x = bhPtr[0];  fh.y = bhPtr[M];
        accZ = wmma4(az, fz, accZ);
        accH = wmma4(ah, fh, accH);
        azPtr += 4;
        ahPtr += 4;
        bzPtr += (size_t)4 * M;
        bhPtr += (size_t)4 * M;
    }

    const float bzv = bz[col];
    const float bhv = bh[col];
    const float wv  = fc_w[col];
#pragma unroll
    for (int i = 0; i < 8; ++i) {
        const int r = rowTile * 16 + i + half * 8;   // uniform across each half-wave
        const float Zv = 1.0f / (1.0f + __expf(-(accZ[i] + bzv)));
        const float Tv = tanhf(accH[i] + bhv);
        const float hv = (1.0f - Zv) * Tv;
        float p = fmaxf(hv, 0.0f) * wv;
        if (r < N) outh[(size_t)r * M + col] = hv;
        // reduce the 16 columns held by this half-wave
#pragma unroll
        for (int off = 8; off > 0; off >>= 1)
            p += __shfl_xor(p, off, 16);
        if (m == 0 && r < N) atomicAdd(&outv[r], p);
    }
}

// ---------------------------------------------------------------------------
// Degree / scatter / init helper kernels
// ---------------------------------------------------------------------------
__global__ void deg_init(float* __restrict__ deg, int n)
{
    int i = blockIdx.x * blockDim.x + threadIdx.x;
    if (i < n) deg[i] = 1.0f;                 // self loop weight
}

__global__ void deg_accum(const int* __restrict__ dst, const float* __restrict__ ew,
                          float* __restrict__ deg, int E)
{
    int e = blockIdx.x * blockDim.x + threadIdx.x;
    if (e < E) atomicAdd(&deg[dst[e]], ew[e]);
}

__global__ void deg_rsqrt(float* __restrict__ deg, int n)
{
    int i = blockIdx.x * blockDim.x + threadIdx.x;
    if (i < n) deg[i] = rsqrtf(deg[i]);       // deg >= 1 (self loops)
}

__global__ void outv_init(const float* __restrict__ fc_b, float* __restrict__ outv, int n)
{
    int i = blockIdx.x * blockDim.x + threadIdx.x;
    if (i < n) outv[i] = fc_b[0];
}

// One wave per edge; each lane handles 4 of 128 features for both matrices.
__global__ __launch_bounds__(256) void scatter_edges(
    const int* __restrict__ src, const int* __restrict__ dst,
    const float* __restrict__ ew, const float* __restrict__ dinv,
    const float* __restrict__ hz, const float* __restrict__ hh,
    float* __restrict__ aggZ, float* __restrict__ aggH, int E)
{
    const long long gid = (long long)blockIdx.x * blockDim.x + threadIdx.x;
    const int e    = (int)(gid >> 5);
    const int lane = (int)(gid & 31);
    if (e >= E) return;

    const int   s   = src[e];
    const int   d   = dst[e];
    const float nrm = dinv[s] * ew[e] * dinv[d];

    const float4 z4 = ((const float4*)(hz + (size_t)s * 128))[lane];
    const float4 h4 = ((const float4*)(hh + (size_t)s * 128))[lane];

    float* az = aggZ + (size_t)d * 128 + lane * 4;
    float* ah = aggH + (size_t)d * 128 + lane * 4;
    atomicAdd(az + 0, z4.x * nrm);
    atomicAdd(az + 1, z4.y * nrm);
    atomicAdd(az + 2, z4.z * nrm);
    atomicAdd(az + 3, z4.w * nrm);
    atomicAdd(ah + 0, h4.x * nrm);
    atomicAdd(ah + 1, h4.y * nrm);
    atomicAdd(ah + 2, h4.z * nrm);
    atomicAdd(ah + 3, h4.w * nrm);
}

// ---------------------------------------------------------------------------
extern "C" void kernel_launch(void* const* d_in, const int* in_sizes, int n_in,
                              void* d_out, int out_size, void* d_ws, size_t ws_size,
                              hipStream_t stream)
{
    const float* x        = (const float*)d_in[0];
    const int*   ei       = (const int*)  d_in[1];
    const float* ew       = (const float*)d_in[2];
    const float* fc0_w    = (const float*)d_in[3];
    const float* fc0_b    = (const float*)d_in[4];
    // d_in[5] = att: softmax over 1 element == 1.0, folds away.
    const float* conv_z_w = (const float*)d_in[6];
    const float* conv_z_b = (const float*)d_in[7];
    const float* lin_z_w  = (const float*)d_in[8];   // (256,128); H0=0 -> top 128 rows
    const float* lin_z_b  = (const float*)d_in[9];
    // d_in[10..13] = R gate: unused because H0 == 0.
    const float* conv_h_w = (const float*)d_in[14];
    const float* conv_h_b = (const float*)d_in[15];
    const float* lin_h_w  = (const float*)d_in[16];
    const float* lin_h_b  = (const float*)d_in[17];
    const float* fc_w     = (const float*)d_in[18];
    const float* fc_b     = (const float*)d_in[19];

    const int N = in_sizes[0] / 128;   // 100000
    const int E = in_sizes[1] / 2;     // 800000
    const int* src = ei;
    const int* dst = ei + E;

    float* ws   = (float*)d_ws;
    float* Xt   = ws;                         // N*256
    float* hz   = Xt   + (size_t)N * 256;     // N*128
    float* hh   = hz   + (size_t)N * 128;     // N*128
    float* deg  = hh   + (size_t)N * 128;     // N   (becomes dinv)
    float* aggZ = deg  + (size_t)N;           // N*128
    float* aggH = aggZ + (size_t)N * 128;     // N*128

    float* outv = (float*)d_out;              // N
    float* outh = outv + N;                   // N*128

    const int rowTiles = (N + 15) / 16;

    // 1) Xt = relu(x @ fc0_w + fc0_b)                 [N,128]@[128,256]
    wmma_gemm_f32<<<dim3(rowTiles, 256 / 64), 128, 0, stream>>>(
        x, fc0_w, fc0_b, Xt, N, 128, 256, ACT_RELU);

    // 2) symmetric-normalization degrees -> dinv
    deg_init <<<(N + 255) / 256, 256, 0, stream>>>(deg, N);
    deg_accum<<<(E + 255) / 256, 256, 0, stream>>>(dst, ew, deg, E);
    deg_rsqrt<<<(N + 255) / 256, 256, 0, stream>>>(deg, N);

    // 3) fused conv GEMMs + bias/self-loop epilogue   [N,256]@[256,128] x2
    conv_dual_gemm<<<dim3(rowTiles, 2), 128, 0, stream>>>(
        Xt, conv_z_w, conv_h_w, conv_z_b, conv_h_b, deg,
        hz, hh, aggZ, aggH, N);

    // 4) edge scatter with symmetric norm
    {
        long long threads = (long long)E * 32;
        scatter_edges<<<(unsigned)((threads + 255) / 256), 256, 0, stream>>>(
            src, dst, ew, deg, hz, hh, aggZ, aggH, E);
    }

    // 5) fused lin GEMMs + gate + head                [N,128]@[128,128] x2
    outv_init<<<(N + 255) / 256, 256, 0, stream>>>(fc_b, outv, N);
    lin_gate_head<<<dim3(rowTiles, 2), 128, 0, stream>>>(
        aggZ, aggH, lin_z_w, lin_h_w, lin_z_b, lin_h_b, fc_w,
        outv, outh, N);
}